// HGNNPConv_31147102831212
// MI455X (gfx1250) — compile-verified
//
#include <hip/hip_runtime.h>

typedef __attribute__((ext_vector_type(2))) float v2f;
typedef __attribute__((ext_vector_type(8))) float v8f;

#define NV 100000
#define NE 50000
#define NNZP 1600000
#define CIN 256
#define COUT 256

// ---------------------------------------------------------------------------
// Wt[n][k] = W[k][n]  (so B fragments load as contiguous b64)
// ---------------------------------------------------------------------------
__global__ __launch_bounds__(256) void hgnn_transpose_w(const float* __restrict__ W,
                                                        float* __restrict__ Wt) {
  int i = blockIdx.x * 256 + threadIdx.x;   // 0..65535
  int k = i >> 8;
  int n = i & 255;
  Wt[(size_t)n * CIN + k] = W[(size_t)k * COUT + n];
}

// ---------------------------------------------------------------------------
// Xt = X * W + bias   via V_WMMA_F32_16X16X4_F32 (one wave per 16x16 C tile)
// A 16x4 frag: lane holds A[m][k + 2*half + {0,1}], m = lane&15, half = lane>>4
// B 4x16 frag: lane holds W[k + 2*half + {0,1}][col0 + n]  (from Wt, contiguous)
// C 16x16:     VGPR r, lane -> (row = half*8 + r, col = n)
// ---------------------------------------------------------------------------
__global__ __launch_bounds__(128) void hgnn_gemm_wmma(const float* __restrict__ X,
                                                      const float* __restrict__ Wt,
                                                      const float* __restrict__ bias,
                                                      float* __restrict__ Xt) {
  const int lane = threadIdx.x & 31;
  const int wave = threadIdx.x >> 5;
  const int tileM = blockIdx.x;                 // 0..6249
  const int tileN = blockIdx.y * 4 + wave;      // 0..15
  const int row0 = tileM * 16;
  const int col0 = tileN * 16;
  const int n = lane & 15;                      // also = m for A frag
  const int half = lane >> 4;

  v8f acc = {};
  const float* Arow = X  + (size_t)(row0 + n) * CIN + 2 * half;
  const float* Brow = Wt + (size_t)(col0 + n) * CIN + 2 * half;

#pragma unroll 8
  for (int k = 0; k < CIN; k += 4) {
    v2f a = *(const v2f*)(Arow + k);            // A[m][k+2h], A[m][k+2h+1]
    v2f b = *(const v2f*)(Brow + k);            // W[k+2h][col0+n], W[k+2h+1][..]
    acc = __builtin_amdgcn_wmma_f32_16x16x4_f32(false, a, false, b,
                                                (short)0, acc, false, false);
  }

  const float bv = bias[col0 + n];
  float* out = Xt + (size_t)(row0 + half * 8) * COUT + col0 + n;
#pragma unroll
  for (int r = 0; r < 8; ++r) {
    out[(size_t)r * COUT] = acc[r] + bv;
  }
}

// ---------------------------------------------------------------------------
__global__ void hgnn_zero4(float4* __restrict__ p, long n4) {
  long i = (long)blockIdx.x * blockDim.x + threadIdx.x;
  if (i < n4) p[i] = make_float4(0.f, 0.f, 0.f, 0.f);
}

// one f32 atomic per pair per side -> segment counts
__global__ void hgnn_count(const int* __restrict__ pv, const int* __restrict__ pe,
                           float* __restrict__ vcnt, float* __restrict__ ecnt) {
  int i = blockIdx.x * blockDim.x + threadIdx.x;
  if (i < NNZP) {
    atomicAdd(&vcnt[pv[i]], 1.0f);
    atomicAdd(&ecnt[pe[i]], 1.0f);
  }
}

// One wave per pair: uniform (scalar) index loads, SADDR-form b128 gathers,
// 8 f32 atomics per lane into the destination segment row.
// If cnt != nullptr the gathered row is pre-scaled by 1/max(cnt[g],1)
// (fuses the edge-mean pass into the e2v gather).
__global__ __launch_bounds__(256) void hgnn_scatter(const float* __restrict__ src,
                                                    const float* __restrict__ cnt,
                                                    const int* __restrict__ gidx,
                                                    const int* __restrict__ sidx,
                                                    float* __restrict__ dst) {
  const int wave = __builtin_amdgcn_readfirstlane((int)(threadIdx.x >> 5));
  const int lane = threadIdx.x & 31;
  const long p = (long)blockIdx.x * 8 + wave;   // wave-uniform pair index
  const int g = gidx[p];                        // -> s_load (uniform)
  const int s = sidx[p];                        // -> s_load (uniform)

  float scale = 1.0f;
  if (cnt != nullptr) scale = 1.0f / fmaxf(cnt[g], 1.0f);   // uniform branch

  const float4* srow = (const float4*)(src + (size_t)g * COUT) + lane * 2;
  float4 x0 = srow[0];
  float4 x1 = srow[1];

  float* d = dst + (size_t)s * COUT + lane * 8;
  atomicAdd(d + 0, x0.x * scale);
  atomicAdd(d + 1, x0.y * scale);
  atomicAdd(d + 2, x0.z * scale);
  atomicAdd(d + 3, x0.w * scale);
  atomicAdd(d + 4, x1.x * scale);
  atomicAdd(d + 5, x1.y * scale);
  atomicAdd(d + 6, x1.z * scale);
  atomicAdd(d + 7, x1.w * scale);
}

// sums[row][c] = relu( sums[row][c] / max(cnt[row],1) )   (in place, float4)
__global__ void hgnn_mean_relu4(float4* __restrict__ sums,
                                const float* __restrict__ cnt, long rows) {
  long i = (long)blockIdx.x * blockDim.x + threadIdx.x;
  if (i >= rows * (COUT / 4)) return;
  long row = i >> 6;                            // 64 float4 per row
  float inv = 1.0f / fmaxf(cnt[row], 1.0f);
  float4 v = sums[i];
  v.x = fmaxf(v.x * inv, 0.0f);
  v.y = fmaxf(v.y * inv, 0.0f);
  v.z = fmaxf(v.z * inv, 0.0f);
  v.w = fmaxf(v.w * inv, 0.0f);
  sums[i] = v;
}

// ---------------------------------------------------------------------------
extern "C" void kernel_launch(void* const* d_in, const int* in_sizes, int n_in,
                              void* d_out, int out_size, void* d_ws, size_t ws_size,
                              hipStream_t stream) {
  const float* X    = (const float*)d_in[0];
  const float* W    = (const float*)d_in[1];
  const float* bias = (const float*)d_in[2];
  const int* pair_v = (const int*)d_in[3];
  const int* pair_e = (const int*)d_in[4];
  float* out = (float*)d_out;

  // workspace layout (floats): Xt | esum | ecnt | vcnt | Wt
  float* ws   = (float*)d_ws;
  float* Xt   = ws;                                   // NV*COUT  = 25,600,000
  float* esum = Xt + (size_t)NV * COUT;               // NE*COUT  = 12,800,000
  float* ecnt = esum + (size_t)NE * COUT;             // NE       =     50,000
  float* vcnt = ecnt + NE;                            // NV       =    100,000
  float* Wt   = vcnt + NV;                            // CIN*COUT =     65,536

  // 0) transpose W for contiguous B-fragment loads
  hgnn_transpose_w<<<(CIN * COUT) / 256, 256, 0, stream>>>(W, Wt);

  // 1) zero accumulators (esum|ecnt|vcnt contiguous) and d_out (vertex sums)
  {
    long nz4 = ((long)NE * COUT + NE + NV) / 4;       // 3,237,500 float4
    hgnn_zero4<<<(int)((nz4 + 255) / 256), 256, 0, stream>>>((float4*)esum, nz4);
    long no4 = (long)NV * COUT / 4;                   // 6,400,000 float4
    hgnn_zero4<<<(int)((no4 + 255) / 256), 256, 0, stream>>>((float4*)out, no4);
  }

  // 2) Xt = X @ W + b  (WMMA f32)
  hgnn_gemm_wmma<<<dim3(NV / 16, 4), 128, 0, stream>>>(X, Wt, bias, Xt);

  // 3) segment counts
  hgnn_count<<<(NNZP + 255) / 256, 256, 0, stream>>>(pair_v, pair_e, vcnt, ecnt);

  // 4) v2e: esum[e] += Xt[v]            (raw sums; no scale on gather)
  hgnn_scatter<<<NNZP / 8, 256, 0, stream>>>(Xt, nullptr, pair_v, pair_e, esum);

  // 5) e2v fused with edge mean: out[v] += esum[e] / max(ecnt[e],1)
  hgnn_scatter<<<NNZP / 8, 256, 0, stream>>>(esum, ecnt, pair_e, pair_v, out);

  // 6) vertex mean + ReLU in place
  {
    long nv4 = (long)NV * COUT / 4;
    hgnn_mean_relu4<<<(int)((nv4 + 255) / 256), 256, 0, stream>>>(
        (float4*)out, vcnt, NV);
  }
}